// QuantumConvLayer_46651934769468
// MI455X (gfx1250) — compile-verified
//
#include <hip/hip_runtime.h>
#include <stdint.h>

// Elementwise transcendental map: out[:,2k]   = cos(q[2k])   * cos(pi*x[:,2k])
//                                 out[:,2k+1] = out[:,2k]    * cos(q[2k+1] + pi*x[:,2k+1])
// Memory-bound (512 MB stream, ~22 us floor at 23.3 TB/s). No matrix structure ->
// WMMA inapplicable; the CDNA5 win is the async-to-LDS prefetch pipeline
// (ASYNCcnt) + non-temporal b128 stores + native v_cos_f32 (revolution units).

typedef float f32x4 __attribute__((ext_vector_type(4)));

#define TPB   256   // threads per block (8 wave32s)
#define DEPTH 8     // async pipeline depth (tiles in flight per wave)
#define ITER  32    // tiles per block; 2048 blocks * 256 * 32 = 2^24 float4 groups

__global__ __launch_bounds__(TPB)
void qconv_async_kernel(const float* __restrict__ x,
                        const float* __restrict__ q,
                        float* __restrict__ out)
{
    __shared__ f32x4 stage[DEPTH * TPB];   // 32 KB staging (of 320 KB/WGP)

    const int      t    = threadIdx.x;
    const uint32_t gb   = (uint32_t)blockIdx.x * (TPB * ITER);  // block's first group

    // Per-thread q slice: group index mod 4 == t mod 4, constant across tiles.
    const f32x4 myq = ((const f32x4*)q)[t & 3];
    const float INV2PI = 0.15915494309189535f;           // v_cos takes revolutions
    const float ce0 = __builtin_amdgcn_cosf(myq.x * INV2PI);   // cos(q[2k])
    const float ce1 = __builtin_amdgcn_cosf(myq.z * INV2PI);   // cos(q[2k+2])
    const float qo0 = myq.y * INV2PI;                          // q[2k+1] / 2pi
    const float qo1 = myq.w * INV2PI;                          // q[2k+3] / 2pi

    const uint64_t sbase   = (uint64_t)(uintptr_t)x;           // SADDR base (GVS mode)
    const uint32_t ldsBase = (uint32_t)(uintptr_t)&stage[0];   // low 32b of flat = LDS offset
    f32x4* __restrict__ out4 = (f32x4*)out;

    // ---- prologue: fill the async pipeline (ASYNCcnt = DEPTH) ----
#pragma unroll
    for (int k = 0; k < DEPTH; ++k) {
        uint32_t voff  = (gb + (uint32_t)k * TPB + (uint32_t)t) * 16u;
        uint32_t laddr = ldsBase + ((uint32_t)k * TPB + (uint32_t)t) * 16u;
        asm volatile("global_load_async_to_lds_b128 %0, %1, %2"
                     :: "v"(laddr), "v"(voff), "s"(sbase) : "memory");
    }

    for (int k = 0; k < ITER; ++k) {
        const int buf = k & (DEPTH - 1);

        if (k < ITER - DEPTH) {
            // 8 outstanding; <=7 guarantees the oldest tile has landed in LDS.
            asm volatile("s_wait_asynccnt 0x7" ::: "memory");
        } else {
            // tail: drain everything
            asm volatile("s_wait_asynccnt 0x0" ::: "memory");
        }

        f32x4 v = stage[buf * TPB + t];   // ds_load_b128

        float ze0 = ce0 * __builtin_amdgcn_cosf(0.5f * v.x);
        float zo0 = ze0 * __builtin_amdgcn_cosf(__builtin_fmaf(0.5f, v.y, qo0));
        float ze1 = ce1 * __builtin_amdgcn_cosf(0.5f * v.z);
        float zo1 = ze1 * __builtin_amdgcn_cosf(__builtin_fmaf(0.5f, v.w, qo1));
        f32x4 r;  r.x = ze0;  r.y = zo0;  r.z = ze1;  r.w = zo1;

        // write-once output: stream past L2 with a non-temporal b128 store
        __builtin_nontemporal_store(r, out4 + (gb + (uint32_t)k * TPB + (uint32_t)t));

        if (k + DEPTH < ITER) {
            // close LDS WAR hazard: our ds_load of this buffer must complete
            // before the async engine may overwrite the slot.
            asm volatile("s_wait_dscnt 0x0" ::: "memory");
            uint32_t voff  = (gb + (uint32_t)(k + DEPTH) * TPB + (uint32_t)t) * 16u;
            uint32_t laddr = ldsBase + ((uint32_t)buf * TPB + (uint32_t)t) * 16u;
            asm volatile("global_load_async_to_lds_b128 %0, %1, %2"
                         :: "v"(laddr), "v"(voff), "s"(sbase) : "memory");
        }
    }
}

extern "C" void kernel_launch(void* const* d_in, const int* in_sizes, int n_in,
                              void* d_out, int out_size, void* d_ws, size_t ws_size,
                              hipStream_t stream) {
    const float* x = (const float*)d_in[0];   // (B, 16) float32
    const float* q = (const float*)d_in[1];   // (16,)  float32
    float* out = (float*)d_out;               // (B, 16) float32

    // out_size = B*16 = 67,108,864 floats -> 2^24 float4 groups
    const long long ngroups = (long long)out_size / 4;
    const int blocks = (int)(ngroups / ((long long)TPB * ITER));   // 2048

    qconv_async_kernel<<<blocks, TPB, 0, stream>>>(x, q, out);
}